// DrosophilaOpticLobeCircuit_59837484368216
// MI455X (gfx1250) — compile-verified
//
#include <hip/hip_runtime.h>
#include <cstdint>

// Problem constants (match reference)
#define N_NEURONS  49000
#define N_PER_TYPE 7000
#define N_TM1      7000
#define N_EDGES    1600000
#define BATCH      8
#define STEPS      100
#define DT         0.1f

// Step-kernel tiling
#define SK_BLOCK 128
#define SK_WAVES (SK_BLOCK / 32)   // 4 waves (wave32)
#define GROUP    32                // targets per wave (one per lane)
#define CAP      1536              // max staged edge-pairs per wave (avg ~1045, sd ~32)

// ---------------------------------------------------------------------------
// CDNA5 async global->LDS copy: each active lane copies 16 bytes.
// Tracked by ASYNCcnt, drained with s_wait_asynccnt.
// ---------------------------------------------------------------------------
__device__ __forceinline__ void async_ld16(void* lds_dst, const void* gsrc) {
  unsigned l = (unsigned)(uintptr_t)lds_dst;                  // low 32 bits of a
  unsigned long long g = (unsigned long long)(uintptr_t)gsrc; //  shared ptr = LDS addr
  asm volatile("global_load_async_to_lds_b128 %0, %1, off"
               :: "v"(l), "v"(g) : "memory");
}

// ============================ CSR build (once per launch) ====================
__global__ void zero_kernel(int* __restrict__ counts) {
  int i = blockIdx.x * blockDim.x + threadIdx.x;
  if (i < N_NEURONS) counts[i] = 0;
}

__global__ void hist_kernel(const int* __restrict__ tgt, int* __restrict__ counts) {
  int e = blockIdx.x * blockDim.x + threadIdx.x;
  if (e < N_EDGES) atomicAdd(&counts[tgt[e]], 1);
}

// Exclusive scan of counts -> row_ptr[0..N] (+ working copy offs), single block.
#define SCAN_T 1024
__global__ void __launch_bounds__(SCAN_T)
scan_kernel(const int* __restrict__ counts, int* __restrict__ row_ptr,
            int* __restrict__ offs) {
  __shared__ int sh[SCAN_T];
  __shared__ int carry_sh;
  const int tid = threadIdx.x;
  if (tid == 0) carry_sh = 0;
  __syncthreads();
  for (int base = 0; base < N_NEURONS; base += SCAN_T) {
    int i = base + tid;
    int x = (i < N_NEURONS) ? counts[i] : 0;
    sh[tid] = x;
    __syncthreads();
    for (int d = 1; d < SCAN_T; d <<= 1) {   // Hillis-Steele inclusive scan
      int t = (tid >= d) ? sh[tid - d] : 0;
      __syncthreads();
      sh[tid] += t;
      __syncthreads();
    }
    int excl = sh[tid] - x + carry_sh;
    if (i < N_NEURONS) { row_ptr[i] = excl; offs[i] = excl; }
    __syncthreads();
    if (tid == SCAN_T - 1) carry_sh += sh[tid];
    __syncthreads();
  }
  if (tid == 0) row_ptr[N_NEURONS] = carry_sh;   // == N_EDGES
}

// Permute edges into target-major order as interleaved {src, weight} pairs;
// fold the (src_type, tgt_type) scale into the weight.
__global__ void scatter_kernel(const int* __restrict__ src, const int* __restrict__ tgt,
                               const float* __restrict__ weights,
                               const float* __restrict__ scale_mat,
                               int* __restrict__ offs, int2* __restrict__ pairs) {
  int e = blockIdx.x * blockDim.x + threadIdx.x;
  if (e >= N_EDGES) return;
  int s = src[e], t = tgt[e];
  unsigned ts = (unsigned)s / N_PER_TYPE;   // mulhi, no real divide
  unsigned tt = (unsigned)t / N_PER_TYPE;
  float w = weights[e] * scale_mat[ts * 7 + tt];
  int p = atomicAdd(&offs[t], 1);
  pairs[p] = make_int2(s, __float_as_int(w));   // one 8B store
}

// ============================ state init ====================================
// v stored neuron-major [N][8]. Tm1 rows = tm1_input in BOTH buffers (frozen
// forever, never rewritten); all other rows start at 0.
__global__ void init_kernel(float* __restrict__ v0, float* __restrict__ v1,
                            const float* __restrict__ tm1) {
  int n = blockIdx.x * blockDim.x + threadIdx.x;
  if (n >= N_NEURONS) return;
  float4 a = make_float4(0.f, 0.f, 0.f, 0.f), b = a;
  if (n < N_TM1) {
    a.x = tm1[0 * N_TM1 + n]; a.y = tm1[1 * N_TM1 + n];
    a.z = tm1[2 * N_TM1 + n]; a.w = tm1[3 * N_TM1 + n];
    b.x = tm1[4 * N_TM1 + n]; b.y = tm1[5 * N_TM1 + n];
    b.z = tm1[6 * N_TM1 + n]; b.w = tm1[7 * N_TM1 + n];
  }
  float4* p0 = reinterpret_cast<float4*>(v0 + (size_t)n * BATCH);
  float4* p1 = reinterpret_cast<float4*>(v1 + (size_t)n * BATCH);
  p0[0] = a; p0[1] = b;
  p1[0] = a; p1[1] = b;
}

// ============================ fused step kernel =============================
// One lane per target neuron. The wave's 32 targets own one contiguous CSR
// edge window of {src,w} pairs, cooperatively async-staged into LDS; lanes
// then accumulate 8 batch sums in registers (no atomics) and apply the
// integrator update into the ping-pong "new" buffer.
__global__ void __launch_bounds__(SK_BLOCK)
step_kernel(const float* __restrict__ v_old, float* __restrict__ v_new,
            const int* __restrict__ row_ptr, const int2* __restrict__ pairs,
            const float* __restrict__ tau_types,
            const float* __restrict__ vrest_types) {
  __shared__ __align__(16) int2 lpair[SK_WAVES][CAP];

  const int lane = threadIdx.x & 31;
  const int wv   = threadIdx.x >> 5;
  const int n0   = (blockIdx.x * SK_WAVES + wv) * GROUP;
  if (n0 >= N_NEURONS) return;                       // wave-uniform
  const int nEnd = (n0 + GROUP < N_NEURONS) ? (n0 + GROUP) : N_NEURONS;

  const int eb   = row_ptr[n0];
  const int ee   = row_ptr[nEnd];
  const int ebA  = eb & ~1;                          // 16B-align window (2 pairs)
  const int lenA = ((ee + 1) & ~1) - ebA;            // (pairs array padded +4)
  const bool use_lds = (lenA <= CAP);                // wave-uniform; ~always true

  if (use_lds) {
    for (int off = lane * 2; off < lenA; off += 64) {   // 16B (2 pairs) per lane
      async_ld16(&lpair[wv][off], pairs + ebA + off);
    }
    asm volatile("s_wait_asynccnt 0x0" ::: "memory");
  }

  const int n = n0 + lane;
  if (n >= N_NEURONS) return;
  const int es = row_ptr[n];
  const int ef = row_ptr[n + 1];

  float a0 = 0.f, a1 = 0.f, a2 = 0.f, a3 = 0.f;
  float a4 = 0.f, a5 = 0.f, a6 = 0.f, a7 = 0.f;
  if (use_lds) {
    for (int e = es; e < ef; ++e) {
      int2  pr = lpair[wv][e - ebA];                 // one ds_load_b64
      int   s  = pr.x;
      float w  = __int_as_float(pr.y);
      const float4* vp = reinterpret_cast<const float4*>(v_old + (size_t)s * BATCH);
      float4 x = vp[0], y = vp[1];                   // 32B contiguous gather
      a0 = fmaf(fmaxf(x.x, 0.f), w, a0);
      a1 = fmaf(fmaxf(x.y, 0.f), w, a1);
      a2 = fmaf(fmaxf(x.z, 0.f), w, a2);
      a3 = fmaf(fmaxf(x.w, 0.f), w, a3);
      a4 = fmaf(fmaxf(y.x, 0.f), w, a4);
      a5 = fmaf(fmaxf(y.y, 0.f), w, a5);
      a6 = fmaf(fmaxf(y.z, 0.f), w, a6);
      a7 = fmaf(fmaxf(y.w, 0.f), w, a7);
    }
  } else {                                           // safety fallback
    for (int e = es; e < ef; ++e) {
      int2  pr = pairs[e];
      int   s  = pr.x;
      float w  = __int_as_float(pr.y);
      const float4* vp = reinterpret_cast<const float4*>(v_old + (size_t)s * BATCH);
      float4 x = vp[0], y = vp[1];
      a0 = fmaf(fmaxf(x.x, 0.f), w, a0);
      a1 = fmaf(fmaxf(x.y, 0.f), w, a1);
      a2 = fmaf(fmaxf(x.z, 0.f), w, a2);
      a3 = fmaf(fmaxf(x.w, 0.f), w, a3);
      a4 = fmaf(fmaxf(y.x, 0.f), w, a4);
      a5 = fmaf(fmaxf(y.y, 0.f), w, a5);
      a6 = fmaf(fmaxf(y.z, 0.f), w, a6);
      a7 = fmaf(fmaxf(y.w, 0.f), w, a7);
    }
  }

  if (n < N_TM1) return;                             // Tm1 frozen, never rewritten
  unsigned t = (unsigned)n / N_PER_TYPE;
  float it = 1.0f / tau_types[t];
  float dr = vrest_types[t];
  const float4* op = reinterpret_cast<const float4*>(v_old + (size_t)n * BATCH);
  float4 u0 = op[0], u1 = op[1];
  u0.x += DT * ((a0 - u0.x + dr) * it);
  u0.y += DT * ((a1 - u0.y + dr) * it);
  u0.z += DT * ((a2 - u0.z + dr) * it);
  u0.w += DT * ((a3 - u0.w + dr) * it);
  u1.x += DT * ((a4 - u1.x + dr) * it);
  u1.y += DT * ((a5 - u1.y + dr) * it);
  u1.z += DT * ((a6 - u1.z + dr) * it);
  u1.w += DT * ((a7 - u1.w + dr) * it);
  float4* np = reinterpret_cast<float4*>(v_new + (size_t)n * BATCH);
  np[0] = u0; np[1] = u1;
}

// ============================ output transpose ==============================
__global__ void out_kernel(const float* __restrict__ v, float* __restrict__ out) {
  int n = blockIdx.x * blockDim.x + threadIdx.x;
  if (n >= N_NEURONS) return;
  const float4* vp = reinterpret_cast<const float4*>(v + (size_t)n * BATCH);
  float4 a = vp[0], b = vp[1];
  out[0 * N_NEURONS + n] = a.x;
  out[1 * N_NEURONS + n] = a.y;
  out[2 * N_NEURONS + n] = a.z;
  out[3 * N_NEURONS + n] = a.w;
  out[4 * N_NEURONS + n] = b.x;
  out[5 * N_NEURONS + n] = b.y;
  out[6 * N_NEURONS + n] = b.z;
  out[7 * N_NEURONS + n] = b.w;
}

extern "C" void kernel_launch(void* const* d_in, const int* in_sizes, int n_in,
                              void* d_out, int out_size, void* d_ws, size_t ws_size,
                              hipStream_t stream) {
  const float* tm1       = (const float*)d_in[0];   // [8][7000]
  const float* weights   = (const float*)d_in[1];   // [1.6M]
  const float* tau_types = (const float*)d_in[2];   // [7]
  const float* vrest     = (const float*)d_in[3];   // [7]
  const float* scale_mat = (const float*)d_in[4];   // [7][7]
  const int*   src       = (const int*)d_in[5];     // [1.6M]
  const int*   tgt       = (const int*)d_in[6];     // [1.6M]
  float* out = (float*)d_out;

  // Workspace layout (4-byte words, each array 16B-aligned):
  //  v0[392000] | v1[392000] | row_ptr[49004] | offs[49004] | counts[49004]
  //  | pairs[1600004 x int2]                         (~16.5 MB total)
  float* basep = (float*)d_ws;
  float* v0      = basep;
  float* v1      = v0 + 392000;
  int*   row_ptr = (int*)(v1 + 392000);
  int*   offs    = row_ptr + 49004;
  int*   counts  = offs + 49004;
  int2*  pairs   = (int2*)(counts + 49004);

  const int T = 256;
  const int blkN = (N_NEURONS + T - 1) / T;     // 192
  const int blkE = (N_EDGES + T - 1) / T;       // 6250
  const int skWaves  = (N_NEURONS + GROUP - 1) / GROUP;          // 1532
  const int skBlocks = (skWaves + SK_WAVES - 1) / SK_WAVES;      // 383

  // ---- CSR-by-target build (amortized over 100 steps) ----
  zero_kernel<<<blkN, T, 0, stream>>>(counts);
  hist_kernel<<<blkE, T, 0, stream>>>(tgt, counts);
  scan_kernel<<<1, SCAN_T, 0, stream>>>(counts, row_ptr, offs);
  scatter_kernel<<<blkE, T, 0, stream>>>(src, tgt, weights, scale_mat, offs, pairs);

  // ---- state init + 100 fused steps (ping-pong v) ----
  init_kernel<<<blkN, T, 0, stream>>>(v0, v1, tm1);
  float* bufs[2] = { v0, v1 };
  for (int s = 0; s < STEPS; ++s) {
    step_kernel<<<skBlocks, SK_BLOCK, 0, stream>>>(
        bufs[s & 1], bufs[(s + 1) & 1], row_ptr, pairs, tau_types, vrest);
  }
  out_kernel<<<blkN, T, 0, stream>>>(bufs[STEPS & 1], out);
}